// SmallObjectYOLOLoss_54468775248445
// MI455X (gfx1250) — compile-verified
//
#include <hip/hip_runtime.h>
#include <hip/hip_bf16.h>

#define EPS_F 1e-7f

typedef __attribute__((ext_vector_type(2))) float v2f;
typedef __attribute__((ext_vector_type(8))) float v8f;

// Stable softplus == jnp.logaddexp(0, x) = max(x,0) + log1p(exp(-|x|))
__device__ __forceinline__ float softplusf(float x) {
    return fmaxf(x, 0.0f) + log1pf(__expf(-fabsf(x)));
}

__device__ __forceinline__ float sigmoidf(float x) {
    return 1.0f / (1.0f + __expf(-x));
}

// ---------------------------------------------------------------------------
// Kernel A: partial sums of softplus over the objectness channels (c = a*6+4)
// of one scale. pred layout: (64, 18, H, W) fp32 row-major. HW4 = H*W/4.
// Each block writes one partial to partials[blockIdx.x].
// Block reduction uses V_WMMA_F32_16X16X4_F32 with A = ones(16x4):
//   D[m][n] = sum_k B[k][n]  -> every row of D holds the 16 column sums of B.
// Chaining 4 WMMAs over 4x64 LDS partials accumulates all 256 values; a
// 16-lane shfl_xor finishes the grand total.
// ---------------------------------------------------------------------------
template <int HW4>
__global__ void __launch_bounds__(256)
obj_partial_kernel(const float* __restrict__ pred, float* __restrict__ partials) {
    constexpr unsigned kB = 64u;
    constexpr unsigned kN4 = kB * 3u * (unsigned)HW4;

    const unsigned tid = threadIdx.x;
    const unsigned gtid = blockIdx.x * blockDim.x + tid;
    const unsigned stride = gridDim.x * blockDim.x;

    float acc = 0.0f;
    for (unsigned e = gtid; e < kN4; e += stride) {
        unsigned b = e / (3u * HW4);                 // image
        unsigned r = e - b * (3u * HW4);
        unsigned a = r / (unsigned)HW4;              // anchor
        unsigned q = r - a * (unsigned)HW4;          // float4 index within channel
        const float4* ch = reinterpret_cast<const float4*>(
            pred + (size_t)(b * 18u + a * 6u + 4u) * (size_t)(HW4 * 4));
        float4 v = ch[q];
        acc += softplusf(v.x) + softplusf(v.y) + softplusf(v.z) + softplusf(v.w);
    }

    __shared__ float sdata[256];
    sdata[tid] = acc;
    __syncthreads();

    if (tid < 32) {  // one fully-active wave32 -> EXEC all 1s for WMMA
        v2f amat;
        amat[0] = 1.0f;
        amat[1] = 1.0f;
        v8f c = {};
#pragma unroll
        for (int chunk = 0; chunk < 4; ++chunk) {
            v2f bmat;
            bmat[0] = sdata[chunk * 64 + tid];
            bmat[1] = sdata[chunk * 64 + 32 + tid];
            // D = ones(16x4) x B(4x16) + C : accumulates column sums of B.
            c = __builtin_amdgcn_wmma_f32_16x16x4_f32(
                /*neg_a=*/false, amat, /*neg_b=*/false, bmat,
                /*c_mod=*/(short)0, c, /*reuse_a=*/false, /*reuse_b=*/false);
        }
        // c[0]: lanes 0-15 hold row M=0 (col sums n=0..15); sum them.
        float v = c[0];
        v += __shfl_xor(v, 1, 32);
        v += __shfl_xor(v, 2, 32);
        v += __shfl_xor(v, 4, 32);
        v += __shfl_xor(v, 8, 32);
        if (tid == 0) partials[blockIdx.x] = v;
    }
}

// ---------------------------------------------------------------------------
// Kernel B: reduce per-scale partials, do the scalar box-loss / anchor math,
// and emit [total, box_loss, obj_loss, cls_loss(=0), total].
// ---------------------------------------------------------------------------
__global__ void __launch_bounds__(256)
finalize_kernel(const float* __restrict__ p0, const float* __restrict__ p1,
                const float* __restrict__ p2, const float* __restrict__ boxes,
                const float* __restrict__ partials,
                int nb0, int nb1, int nb2, float* __restrict__ out) {
    __shared__ float red[256];
    __shared__ float ssum[3];
    const int tid = threadIdx.x;

    int offs[4];
    offs[0] = 0; offs[1] = nb0; offs[2] = nb0 + nb1; offs[3] = nb0 + nb1 + nb2;

    for (int seg = 0; seg < 3; ++seg) {
        float a = 0.0f;
        for (int i = offs[seg] + tid; i < offs[seg + 1]; i += 256) a += partials[i];
        red[tid] = a;
        __syncthreads();
        for (int s = 128; s > 0; s >>= 1) {
            if (tid < s) red[tid] += red[tid + s];
            __syncthreads();
        }
        if (tid == 0) ssum[seg] = red[0];
        __syncthreads();
    }

    if (tid != 0) return;

    // Anchors (already divided by strides 8/16/32 in the reference).
    const float anc[3][3][2] = {
        {{10.f/8.f, 13.f/8.f}, {16.f/8.f, 30.f/8.f}, {33.f/8.f, 23.f/8.f}},
        {{30.f/16.f, 61.f/16.f}, {62.f/16.f, 45.f/16.f}, {59.f/16.f, 119.f/16.f}},
        {{116.f/32.f, 90.f/32.f}, {156.f/32.f, 198.f/32.f}, {373.f/32.f, 326.f/32.f}},
    };
    const int dims[3] = {160, 80, 40};
    const float* preds[3] = {p0, p1, p2};

    // Only the last image (b = 63) survives in the reference (indices overwritten).
    const float tbx = boxes[63 * 4 + 0];
    const float tby = boxes[63 * 4 + 1];
    const float tbw = boxes[63 * 4 + 2];
    const float tbh = boxes[63 * 4 + 3];

    float box_sum = 0.0f;
    float obj_sum = 0.0f;

    for (int s = 0; s < 3; ++s) {
        const int W = dims[s], H = dims[s];
        const float gx = tbx * (float)W;
        const float gy = tby * (float)H;
        const float gw = tbw * (float)W;
        const float gh = tbh * (float)H;
        int gi = (int)gx; gi = gi < 0 ? 0 : (gi > W - 1 ? W - 1 : gi);
        int gj = (int)gy; gj = gj < 0 ? 0 : (gj > H - 1 ? H - 1 : gj);

        // best anchor: max IoU of zero-centered wh boxes (strict >, first wins ties)
        int best = 0;
        float bestv = -1.0f;
        for (int a = 0; a < 3; ++a) {
            const float aw = anc[s][a][0], ah = anc[s][a][1];
            const float inter = fminf(gw, aw) * fminf(gh, ah);
            const float uni = gw * gh + aw * ah - inter + EPS_F;
            const float r = inter / uni;
            if (r > bestv) { bestv = r; best = a; }
        }

        const size_t HW = (size_t)H * (size_t)W;
        const size_t idx0 = ((size_t)(63 * 18 + best * 6)) * HW + (size_t)gj * W + (size_t)gi;
        const float* pr = preds[s];
        float ps0 = pr[idx0 + 0 * HW];
        float ps1 = pr[idx0 + 1 * HW];
        float ps2 = pr[idx0 + 2 * HW];
        float ps3 = pr[idx0 + 3 * HW];
        float ps4 = pr[idx0 + 4 * HW];   // objectness logit at the positive cell

        // pred box (sigmoid xy, exp wh * anchor / 2), target = normalized box.
        const float px = sigmoidf(ps0), py = sigmoidf(ps1);
        const float hx = __expf(ps2) * anc[s][best][0] * 0.5f;
        const float hy = __expf(ps3) * anc[s][best][1] * 0.5f;
        const float b1x1 = px - hx, b1y1 = py - hy, b1x2 = px + hx, b1y2 = py + hy;
        const float b2x1 = tbx - tbw * 0.5f, b2y1 = tby - tbh * 0.5f;
        const float b2x2 = tbx + tbw * 0.5f, b2y2 = tby + tbh * 0.5f;

        const float w1 = b1x2 - b1x1, h1 = fmaxf(b1y2 - b1y1, EPS_F);
        const float w2 = b2x2 - b2x1, h2 = fmaxf(b2y2 - b2y1, EPS_F);
        const float iw = fmaxf(fminf(b1x2, b2x2) - fmaxf(b1x1, b2x1), 0.0f);
        const float ih = fmaxf(fminf(b1y2, b2y2) - fmaxf(b1y1, b2y1), 0.0f);
        const float inter = iw * ih;
        const float uni = w1 * h1 + w2 * h2 - inter + EPS_F;
        const float iou = inter / uni;

        const float cw = fmaxf(b1x2, b2x2) - fminf(b1x1, b2x1);
        const float ch = fmaxf(b1y2, b2y2) - fminf(b1y1, b2y1);
        const float dx = b2x1 + b2x2 - b1x1 - b1x2;
        const float dy = b2y1 + b2y2 - b1y1 - b1y2;
        const float rho2 = (dx * dx + dy * dy) * 0.25f;
        const float c2 = cw * cw + ch * ch + EPS_F;
        const float rho_w2 = (w2 - w1) * (w2 - w1) / (cw * cw + EPS_F);
        const float rho_h2 = (h2 - h1) * (h2 - h1) / (ch * ch + EPS_F);
        const float eiou = iou - (rho2 / c2 + rho_w2 + rho_h2);

        box_sum += sqrtf(iou) * (1.0f - eiou);   // iou^GAMMA * (1 - eiou), GAMMA=0.5

        const float N = 64.0f * 3.0f * (float)H * (float)W;
        obj_sum += (ssum[s] - ps4) / N;          // mean(softplus(x) - x*t)
    }

    const float cls_sum = 0.0f;                  // nc == 1 -> inactive
    const float total = 0.1f * box_sum + 1.0f * obj_sum + 0.5f * cls_sum;

    out[0] = total;     // returned scalar
    out[1] = box_sum;   // loss_items[0]
    out[2] = obj_sum;   // loss_items[1]
    out[3] = cls_sum;   // loss_items[2]
    out[4] = total;     // loss_items[3]
}

// ---------------------------------------------------------------------------
extern "C" void kernel_launch(void* const* d_in, const int* in_sizes, int n_in,
                              void* d_out, int out_size, void* d_ws, size_t ws_size,
                              hipStream_t stream) {
    const float* pred0 = (const float*)d_in[0];  // (64,18,160,160)
    const float* pred1 = (const float*)d_in[1];  // (64,18, 80, 80)
    const float* pred2 = (const float*)d_in[2];  // (64,18, 40, 40)
    const float* boxes = (const float*)d_in[3];  // (64,4)
    // d_in[4] = labels, unused (nc == 1)

    float* ws = (float*)d_ws;
    constexpr int NB0 = 4096, NB1 = 1024, NB2 = 256;

    obj_partial_kernel<160 * 160 / 4><<<NB0, 256, 0, stream>>>(pred0, ws + 0);
    obj_partial_kernel< 80 *  80 / 4><<<NB1, 256, 0, stream>>>(pred1, ws + NB0);
    obj_partial_kernel< 40 *  40 / 4><<<NB2, 256, 0, stream>>>(pred2, ws + NB0 + NB1);

    finalize_kernel<<<1, 256, 0, stream>>>(pred0, pred1, pred2, boxes, ws,
                                           NB0, NB1, NB2, (float*)d_out);
}